// SSIMLoss_36739150250609
// MI455X (gfx1250) — compile-verified
//
#include <hip/hip_runtime.h>
#include <stdint.h>

// Problem constants (from reference setup_inputs)
#define NN 4
#define DD 64
#define HH 256
#define WW 256

// Tile configuration
#define TW 64          // output tile width
#define TH 16          // output tile height
#define DCHUNK 16      // depth slices per block
#define RW (TW + 2)    // 66: raw tile width (halo)
#define RH (TH + 2)    // 18: raw tile height (halo)
#define NTHREADS 256

__constant__ const float kC1 = 0.01f * 0.01f;
__constant__ const float kC2 = 0.03f * 0.03f;

// ---- CDNA5 async global -> LDS copy (ASYNCcnt path) -----------------------
// VDST = VGPR holding byte offset into the workgroup's LDS allocation,
// VADDR = 64-bit global address. Per-lane predicated via EXEC.
__device__ __forceinline__ uint32_t lds_byte_off(const void* p) {
  // Generic LDS pointers on gfx12xx are {shared_aperture_hi32 | lds_offset_lo32};
  // the low 32 bits are the wave-relative LDS byte address.
  return (uint32_t)(uintptr_t)p;
}

__device__ __forceinline__ void async_copy_b32(uint32_t lds_off, const float* g) {
  asm volatile("global_load_async_to_lds_b32 %0, %1, off"
               :: "v"(lds_off), "v"((uint64_t)(uintptr_t)g)
               : "memory");
}

__device__ __forceinline__ void wait_async_zero() {
  asm volatile("s_wait_asynccnt 0" ::: "memory");
}
// ---------------------------------------------------------------------------

__global__ void ssim_zero_acc(double* acc) {
  acc[0] = 0.0;
  acc[1] = 0.0;
}

__global__ __launch_bounds__(NTHREADS)
void ssim_main(const float* __restrict__ x, const float* __restrict__ y,
               const float* __restrict__ msk, double* __restrict__ acc) {
  // Double-buffered raw haloed slices (async fill target)
  __shared__ float smx[2][RH * RW];
  __shared__ float smy[2][RH * RW];
  __shared__ float ring[5][3][TH * TW];  // 2D 3x3 partial sums, 3-slice ring
  __shared__ float red[NTHREADS];

  const int tid = threadIdx.x;
  const int tileW = blockIdx.x & 3;   // 4 tiles across W
  const int tileH = blockIdx.x >> 2;  // 16 tiles across H
  const int d0 = blockIdx.y * DCHUNK;
  const int n = blockIdx.z;
  const int w0 = tileW * TW;
  const int h0 = tileH * TH;
  const size_t nbase = (size_t)n * DD * HH * WW;
  const float inv27 = 1.0f / 27.0f;

  float p_ssim = 0.0f;
  float p_mask = 0.0f;

  // Issue async fill for slice z into buffer `buf` (caller guarantees z valid).
  auto fill_slice = [&](int z, int buf) {
    const size_t zbase = nbase + (size_t)z * HH * WW;
    for (int i = tid; i < RH * RW; i += NTHREADS) {
      const int tw = i % RW;
      const int th = i / RW;
      const int w = w0 - 1 + tw;
      const int h = h0 - 1 + th;
      if ((unsigned)w < WW && (unsigned)h < HH) {
        const size_t g = zbase + (size_t)h * WW + (size_t)w;
        async_copy_b32(lds_byte_off(&smx[buf][i]), x + g);
        async_copy_b32(lds_byte_off(&smy[buf][i]), y + g);
      } else {
        smx[buf][i] = 0.0f;  // SAME zero-padding
        smy[buf][i] = 0.0f;
      }
    }
  };

  // Prologue: start fill of the first slice (z = d0-1) into buffer 0.
  if (d0 - 1 >= 0) fill_slice(d0 - 1, 0);

  for (int z = d0 - 1; z <= d0 + DCHUNK; ++z) {
    const int slot = (z - (d0 - 1)) % 3;
    const int cur = (z - (d0 - 1)) & 1;
    const bool zvalid = (z >= 0) && (z < DD);

    // Slice z's copies (issued one iteration ago) must have landed for every
    // wave before anyone reads them.
    wait_async_zero();
    __syncthreads();

    // Kick off slice z+1's fill into the other buffer; it overlaps with the
    // compute below. (Buffer cur^1 was last read before the barrier above.)
    const int zn = z + 1;
    if (zn <= d0 + DCHUNK && zn >= 0 && zn < DD) fill_slice(zn, cur ^ 1);

    // Stage 2: 2D 3x3 partial sums for the 5 quantities into ring[*][slot].
    if (zvalid) {
      for (int i = tid; i < TH * TW; i += NTHREADS) {
        const int tw = i % TW;
        const int th = i / TW;
        float ax = 0.f, ay = 0.f, axx = 0.f, ayy = 0.f, axy = 0.f;
#pragma unroll
        for (int dh = 0; dh < 3; ++dh) {
#pragma unroll
          for (int dw = 0; dw < 3; ++dw) {
            const int r = (th + dh) * RW + (tw + dw);
            const float sx = smx[cur][r];
            const float sy = smy[cur][r];
            ax += sx;
            ay += sy;
            axx = fmaf(sx, sx, axx);
            ayy = fmaf(sy, sy, ayy);
            axy = fmaf(sx, sy, axy);
          }
        }
        ring[0][slot][i] = ax;
        ring[1][slot][i] = ay;
        ring[2][slot][i] = axx;
        ring[3][slot][i] = ayy;
        ring[4][slot][i] = axy;
      }
    } else {
      for (int i = tid; i < TH * TW; i += NTHREADS) {
        ring[0][slot][i] = 0.f;
        ring[1][slot][i] = 0.f;
        ring[2][slot][i] = 0.f;
        ring[3][slot][i] = 0.f;
        ring[4][slot][i] = 0.f;
      }
    }
    __syncthreads();

    // Stage 3: emit output slice d = z-1 from ring slots (d-1, d, d+1).
    const int d = z - 1;
    if (d >= d0 && d < d0 + DCHUNK) {
      const int s0 = (d - d0) % 3;      // slot of slice d-1
      const int s1 = (d - d0 + 1) % 3;  // slot of slice d
      const int s2 = slot;              // slot of slice d+1
      const size_t dbase = nbase + (size_t)d * HH * WW;
      for (int i = tid; i < TH * TW; i += NTHREADS) {
        const int tw = i % TW;
        const int th = i / TW;
        const float mux = (ring[0][s0][i] + ring[0][s1][i] + ring[0][s2][i]) * inv27;
        const float muy = (ring[1][s0][i] + ring[1][s1][i] + ring[1][s2][i]) * inv27;
        const float mxx = (ring[2][s0][i] + ring[2][s1][i] + ring[2][s2][i]) * inv27;
        const float myy = (ring[3][s0][i] + ring[3][s1][i] + ring[3][s2][i]) * inv27;
        const float mxy = (ring[4][s0][i] + ring[4][s1][i] + ring[4][s2][i]) * inv27;
        const float sig_x = mxx - mux * mux;
        const float sig_y = myy - muy * muy;
        const float sig_xy = mxy - mux * muy;
        const float num = (2.0f * mux * muy + kC1) * (2.0f * sig_xy + kC2);
        const float den = (mux * mux + muy * muy + kC1) * (sig_x + sig_y + kC2);
        const float ssim = num / den;
        const float mk = msk[dbase + (size_t)(h0 + th) * WW + (size_t)(w0 + tw)];
        p_ssim = fmaf(ssim, mk, p_ssim);
        p_mask += mk;
      }
    }
    // The wait+barrier at the top of the next iteration separates these ring
    // reads from the next ring-slot overwrite and orders buffer reuse.
  }

  // Block reduction of the two partials, then fp64 atomics into acc.
  red[tid] = p_ssim;
  __syncthreads();
  for (int s = NTHREADS / 2; s > 0; s >>= 1) {
    if (tid < s) red[tid] += red[tid + s];
    __syncthreads();
  }
  const float block_ssim = red[0];
  __syncthreads();
  red[tid] = p_mask;
  __syncthreads();
  for (int s = NTHREADS / 2; s > 0; s >>= 1) {
    if (tid < s) red[tid] += red[tid + s];
    __syncthreads();
  }
  if (tid == 0) {
    atomicAdd(&acc[0], (double)block_ssim);
    atomicAdd(&acc[1], (double)red[0]);
  }
}

__global__ void ssim_finalize(const double* __restrict__ acc, float* __restrict__ out) {
  const double s = acc[0] / (acc[1] + 1.0e-7);
  out[0] = 1.0f - (float)s;
}

extern "C" void kernel_launch(void* const* d_in, const int* in_sizes, int n_in,
                              void* d_out, int out_size, void* d_ws, size_t ws_size,
                              hipStream_t stream) {
  (void)in_sizes; (void)n_in; (void)out_size; (void)ws_size;
  const float* x = (const float*)d_in[0];   // struct
  const float* y = (const float*)d_in[1];   // gt_struct
  const float* m = (const float*)d_in[2];   // masked
  // d_in[3] is the constant ones/27 kernel; folded into inv27.
  double* acc = (double*)d_ws;              // [ssim_sum, mask_sum]
  float* out = (float*)d_out;

  ssim_zero_acc<<<1, 1, 0, stream>>>(acc);
  dim3 grid(64 /* 4 W-tiles * 16 H-tiles */, DD / DCHUNK, NN);
  ssim_main<<<grid, NTHREADS, 0, stream>>>(x, y, m, acc);
  ssim_finalize<<<1, 1, 0, stream>>>(acc, out);
}